// LocalPathEncoderRobustAdvancedTemporal_53377853555194
// MI455X (gfx1250) — compile-verified
//
#include <hip/hip_runtime.h>

#define BATCH 2048
#define LPOS  64
#define DIM   128
#define EPSF  1e-6f
#define HP    132   // padded row stride for H in LDS (floats)
#define WP    132   // padded row stride for W2 in LDS (floats)

typedef __attribute__((ext_vector_type(2))) float v2f;
typedef __attribute__((ext_vector_type(8))) float v8f;

__launch_bounds__(256)
__global__ void path_encoder_kernel(
    const int*   __restrict__ src_ids, const int*   __restrict__ dst_ids,
    const int*   __restrict__ src_nid, const int*   __restrict__ dst_nid,
    const float* __restrict__ cur_times,
    const float* __restrict__ src_t,   const float* __restrict__ dst_t,
    const float* __restrict__ W1, const float* __restrict__ b1,
    const float* __restrict__ W2, const float* __restrict__ b2,
    float* __restrict__ out)
{
    __shared__ int   s_ids[2][LPOS];
    __shared__ float s_t[2][LPOS];
    __shared__ int   s_rank[2][LPOS];
    __shared__ float s_avg[2][LPOS];
    __shared__ float s_rec[2][LPOS];
    __shared__ float s_feat[2][LPOS][8];
    __shared__ float sW1[DIM], sb1[DIM], sb2[DIM];
    __shared__ float sW2[DIM * WP];        // 67.5 KB
    __shared__ float sH[2][LPOS][HP];      // 67.5 KB

    const int b   = blockIdx.x;
    const int tid = threadIdx.x;

    // ---- stage parameters into LDS ----
    if (tid < DIM) { sW1[tid] = W1[tid]; sb1[tid] = b1[tid]; sb2[tid] = b2[tid]; }
    for (int idx = tid; idx < DIM * DIM; idx += 256)
        sW2[(idx >> 7) * WP + (idx & 127)] = W2[idx];

    // ---- stage per-sample ids / times ----
    if (tid < 128) {
        int side = tid >> 6, i = tid & 63;
        const int*   idp = side ? dst_ids : src_ids;
        const float* tp  = side ? dst_t   : src_t;
        s_ids[side][i] = idp[b * LPOS + i];
        s_t[side][i]   = tp[b * LPOS + i];
    }
    __syncthreads();

    const float cur_t = cur_times[b];
    const int   sid   = src_nid[b];
    const int   did   = dst_nid[b];

    const int side = tid >> 6, i = tid & 63;   // meaningful for tid < 128
    int   myid = 0; float myt = 0.f;
    int   n = 0, rank = 0;
    float tmax = 0.f, tmin = 0.f;
    float myavg = 0.f, myrec = 0.f;

    // ---- pass A: group size, extremes, stable rank ----
    if (tid < 128) {
        myid = s_ids[side][i];
        myt  = s_t[side][i];
        tmax = myt; tmin = myt;
        for (int j = 0; j < LPOS; ++j) {
            if (s_ids[side][j] == myid) {
                float tj = s_t[side][j];
                ++n;
                tmax = fmaxf(tmax, tj);
                tmin = fminf(tmin, tj);
                if (tj < myt || (tj == myt && j < i)) ++rank;
            }
        }
        s_rank[side][i] = rank;
    }
    __syncthreads();

    // ---- pass B: median-split time, avg/recent inter-arrival ----
    if (tid < 128) {
        int split = n >> 1;
        float tsplit = 0.f;
        for (int j = 0; j < LPOS; ++j)
            if (s_ids[side][j] == myid && s_rank[side][j] == split)
                tsplit += s_t[side][j];
        float avg = (n > 1) ? (tmax - tmin) / fmaxf((float)n - 1.f, 1.f) : 0.f;
        float den = fmaxf((float)(n - split) - 1.f, 1.f);
        float rec = (n >= 4) ? (tmax - tsplit) / den : 0.f;
        if (myid == 0) { avg = 0.f; rec = 0.f; }
        s_avg[side][i] = avg; s_rec[side][i] = rec;
        myavg = avg; myrec = rec;
    }
    __syncthreads();

    // ---- pass C: cross-side features ----
    if (tid < 128) {
        int   oside = side ^ 1;
        int   other_node = side ? sid : did;
        float c_self = (float)n;
        float c_other = 0.f, iat_o = 0.f, riat_o = 0.f;
        int   li = -1; bool first = false;
        for (int j = 0; j < LPOS; ++j) {
            if (s_ids[oside][j] == myid) {
                c_other += 1.f;
                if (myid != 0) {
                    li = j;                       // last occurrence
                    if (!first) {                 // first occurrence
                        first = true;
                        iat_o  = s_avg[oside][j];
                        riat_o = s_rec[oside][j];
                    }
                }
            }
        }
        float is_other  = (myid == other_node) ? 1.f : 0.f;
        float connects  = (c_other > 0.f) ? 1.f : 0.f;
        float freq_asym = (c_other > 0.f) ? c_self / (c_other + EPSF) : 0.f;
        float rcy_self  = cur_t - myt;
        float lastt     = (li >= 0) ? s_t[oside][li] : 0.f;
        float rcy_other = cur_t - lastt;
        float temp_asym = (rcy_self > EPSF) ? rcy_other / (rcy_self + EPSF) : 0.f;
        float iat_asym  = (iat_o  > EPSF) ? myavg / (iat_o  + EPSF) : 0.f;
        float riat_asym = (riat_o > EPSF) ? myrec / (riat_o + EPSF) : 0.f;
        float fv[8];
        fv[0] = side ? c_other : c_self;
        fv[1] = side ? c_self  : c_other;
        fv[2] = is_other; fv[3] = connects; fv[4] = freq_asym;
        fv[5] = temp_asym; fv[6] = iat_asym; fv[7] = riat_asym;
        if (myid == 0)
            for (int f = 0; f < 8; ++f) fv[f] = 0.f;
        for (int f = 0; f < 8; ++f) s_feat[side][i][f] = fv[f];
    }
    __syncthreads();

    // ---- H = sum_f relu(feat_f * W1 + b1)  -> LDS (padded) ----
    for (int rep = 0; rep < 64; ++rep) {
        int idx = tid + (rep << 8);        // 0..16383
        int hs = idx >> 13;
        int r  = (idx >> 7) & 63;
        int d  = idx & 127;
        float w = sW1[d], bb = sb1[d];
        float acc = 0.f;
        #pragma unroll
        for (int f = 0; f < 8; ++f)
            acc += fmaxf(fmaf(s_feat[hs][r][f], w, bb), 0.f);
        sH[hs][r][d] = acc;
    }
    __syncthreads();

    // ---- GEMM: Y[64,128] = H[64,128] @ W2[128,128] per side, via f32 WMMA ----
    const int wave = tid >> 5;
    const int lane = tid & 31;
    const int l16  = lane & 15;
    const int hi   = lane >> 4;            // half-wave selects K pair (0,1) vs (2,3)
    const float* Hflat = &sH[0][0][0];

    for (int q = 0; q < 8; ++q) {          // uniform across all 8 waves
        int tt = wave + (q << 3);          // 0..63 tiles
        int sg = tt >> 5;                  // side: 0=src, 1=dst
        int mt = (tt >> 3) & 3;            // m-tile (16 rows)
        int nt = tt & 7;                   // n-tile (16 cols)
        int n0 = nt << 4;

        const float* Arow = Hflat + (size_t)(sg * LPOS + (mt << 4) + l16) * HP + (hi << 1);
        const float* Bcol = &sW2[(hi << 1) * WP + n0 + l16];

        v8f acc = {0.f, 0.f, 0.f, 0.f, 0.f, 0.f, 0.f, 0.f};
        for (int k = 0; k < DIM; k += 4) {
            v2f a = *(const v2f*)(Arow + k);       // A[M=l16][k+2*hi .. +1]
            v2f bb;
            bb.x = Bcol[k * WP];                   // B[k+2*hi  ][n0+l16]
            bb.y = Bcol[k * WP + WP];              // B[k+2*hi+1][n0+l16]
            acc = __builtin_amdgcn_wmma_f32_16x16x4_f32(
                false, a, false, bb, (short)0, acc, false, false);
        }

        int row0 = (mt << 4) + (hi << 3);          // C layout: vgpr j -> row row0+j
        int col  = n0 + l16;
        float badd = 8.f * sb2[col];
        float* op = out + (size_t)sg * (size_t)(BATCH * LPOS * DIM)
                        + ((size_t)b * LPOS + row0) * DIM + col;
        #pragma unroll
        for (int j = 0; j < 8; ++j)
            op[(size_t)j * DIM] = acc[j] + badd;
    }
}

extern "C" void kernel_launch(void* const* d_in, const int* in_sizes, int n_in,
                              void* d_out, int out_size, void* d_ws, size_t ws_size,
                              hipStream_t stream) {
    (void)in_sizes; (void)n_in; (void)out_size; (void)d_ws; (void)ws_size;
    const int*   src_ids = (const int*)d_in[0];
    const int*   dst_ids = (const int*)d_in[1];
    const int*   src_nid = (const int*)d_in[2];
    const int*   dst_nid = (const int*)d_in[3];
    const float* times   = (const float*)d_in[4];
    const float* src_t   = (const float*)d_in[5];
    const float* dst_t   = (const float*)d_in[6];
    const float* W1      = (const float*)d_in[7];
    const float* b1      = (const float*)d_in[8];
    const float* W2      = (const float*)d_in[9];
    const float* b2      = (const float*)d_in[10];
    path_encoder_kernel<<<BATCH, 256, 0, stream>>>(
        src_ids, dst_ids, src_nid, dst_nid, times, src_t, dst_t,
        W1, b1, W2, b2, (float*)d_out);
}